// SAGE_SimWeighted_14448269983760
// MI455X (gfx1250) — compile-verified
//
#include <hip/hip_runtime.h>
#include <math.h>

#define N_NODES 50000
#define N_EDGES 1600000
#define DIN 128
#define DH 128
#define DOUT 64
#define INV_TAU 2.0f
#define EPSF 1e-12f
#define BN_EPSF 1e-5f

typedef float v2f __attribute__((ext_vector_type(2)));
typedef float v8f __attribute__((ext_vector_type(8)));

// ---- monotonic float <-> uint encoding for atomic segment-max ----
__device__ __forceinline__ unsigned f2ord(float f) {
  unsigned u = __float_as_uint(f);
  return (u & 0x80000000u) ? ~u : (u | 0x80000000u);
}
__device__ __forceinline__ float ord2f(unsigned u) {
  return (u & 0x80000000u) ? __uint_as_float(u ^ 0x80000000u)
                           : __uint_as_float(~u);
}

__device__ __forceinline__ float waveReduceSum(float v) {
#pragma unroll
  for (int off = 16; off > 0; off >>= 1) v += __shfl_xor(v, off, 32);
  return v;
}

// ---------------- init segment state ----------------
__global__ void init_seg_kernel(unsigned* __restrict__ mrep,
                                float* __restrict__ denom, int n) {
  int i = blockIdx.x * blockDim.x + threadIdx.x;
  if (i < n) { mrep[i] = 0u; denom[i] = 0.0f; }
}

// ---------------- 1/(||x_row|| + eps), one wave32 per node ----------------
__global__ void row_invnorm_kernel(const float* __restrict__ x,
                                   float* __restrict__ inv, int n) {
  int wid  = (blockIdx.x * blockDim.x + threadIdx.x) >> 5;
  int lane = threadIdx.x & 31;
  if (wid >= n) return;
  const float4* row = (const float4*)(x + (size_t)wid * DIN);
  float4 v = row[lane];
  float s = v.x * v.x + v.y * v.y + v.z * v.z + v.w * v.w;
  s = waveReduceSum(s);
  if (lane == 0) inv[wid] = 1.0f / (sqrtf(s) + EPSF);
}

// ---------------- per-edge logits + atomic segment-max, wave32/edge ----------
__global__ void edge_logits_kernel(const float* __restrict__ x,
                                   const long long* __restrict__ src,
                                   const long long* __restrict__ dst,
                                   const float* __restrict__ inv,
                                   float* __restrict__ logits,
                                   unsigned* __restrict__ mrep, int ne) {
  int e    = (blockIdx.x * blockDim.x + threadIdx.x) >> 5;
  int lane = threadIdx.x & 31;
  if (e >= ne) return;
  int s = (int)src[e], d = (int)dst[e];
  const float4* rs = (const float4*)(x + (size_t)s * DIN);
  const float4* rd = (const float4*)(x + (size_t)d * DIN);
  float4 a = rs[lane], b = rd[lane];
  float dot = a.x * b.x + a.y * b.y + a.z * b.z + a.w * b.w;
  dot = waveReduceSum(dot);
  if (lane == 0) {
    float l = dot * inv[s] * inv[d] * INV_TAU;
    logits[e] = l;
    atomicMax(mrep + d, f2ord(l));
  }
}

// ---------------- e = exp(logit - max[dst]); denom[dst] += e ----------------
__global__ void edge_exp_kernel(const float* __restrict__ logits,
                                const long long* __restrict__ dst,
                                const unsigned* __restrict__ mrep,
                                float* __restrict__ w,
                                float* __restrict__ denom, int ne) {
  int e = blockIdx.x * blockDim.x + threadIdx.x;
  if (e >= ne) return;
  int d = (int)dst[e];
  float ev = __expf(logits[e] - ord2f(mrep[d]));
  w[e] = ev;
  atomicAdd(denom + d, ev);
}

// ---------------- w = e / (denom[dst] + 1e-16) ----------------
__global__ void edge_norm_kernel(float* __restrict__ w,
                                 const long long* __restrict__ dst,
                                 const float* __restrict__ denom, int ne) {
  int e = blockIdx.x * blockDim.x + threadIdx.x;
  if (e >= ne) return;
  w[e] = w[e] / (denom[(int)dst[e]] + 1e-16f);
}

// ---------------- C[M,NC] = A[M,K] @ W[K,NC] (+ b1 + b2), WMMA f32 ----------
// One wave32 per 16x16 output tile; K stepped by 4 with V_WMMA_F32_16X16X4_F32.
// Compile-time K/NC => all loop loads/stores use one base VGPR pair plus an
// immediate ioffset (no per-iteration 64-bit address math competing with WMMA).
// A operand layout (ISA 7.12.2): lanes 0-15 feed K=k+{0,1}, lanes 16-31 feed
// K=k+{2,3}; the two K values per lane are contiguous -> single b64 load.
template <int K, int NC>
__global__ void gemm_wmma_kernel(const float* __restrict__ A,
                                 const float* __restrict__ W,
                                 const float* __restrict__ b1,
                                 const float* __restrict__ b2,
                                 float* __restrict__ C, int M) {
  int waveId = (blockIdx.x * blockDim.x + threadIdx.x) >> 5;
  constexpr int NT = NC >> 4;
  int totalTiles = (M >> 4) * NT;
  if (waveId >= totalTiles) return;
  int mt = waveId / NT, nt = waveId - mt * NT;
  int lane = threadIdx.x & 31;
  int half = lane >> 4, lid = lane & 15;
  int col = (nt << 4) + lid;  // B/C column this lane holds

  // per-lane bases (computed once)
  const float* arow = A + (size_t)((mt << 4) + lid) * K + (half << 1);
  const float* wcol = W + (size_t)(half << 1) * NC + col;

  v8f acc = {};
#pragma unroll
  for (int k = 0; k < K; k += 4) {
    v2f a = *(const v2f*)(arow + k);       // global_load_b64, ioffset = k*4
    v2f b;
    b.x = wcol[(size_t)k * NC];            // global_load_b32, ioffset = k*NC*4
    b.y = wcol[(size_t)k * NC + NC];       // ioffset = (k+1)*NC*4
    acc = __builtin_amdgcn_wmma_f32_16x16x4_f32(
        /*neg_a=*/false, a, /*neg_b=*/false, b,
        /*c_mod=*/(short)0, acc, /*reuse_a=*/false, /*reuse_b=*/false);
  }

  float bias = 0.0f;
  if (b1) bias += b1[col];
  if (b2) bias += b2[col];

  // C layout: VGPR v -> row mbase+v (lanes 0-15: M=v, lanes 16-31: M=v+8)
  int mbase = (mt << 4) + (half << 3);
  float* crow = C + (size_t)mbase * NC + col;
#pragma unroll
  for (int v = 0; v < 8; ++v)
    crow[(size_t)v * NC] = acc[v] + bias;  // global_store_b32, imm offsets
}

// ---------------- out[dst] += p[src] * w, float4 chunks ----------------
template <int D>
__global__ void aggregate_kernel(float* __restrict__ out,
                                 const float* __restrict__ p,
                                 const float* __restrict__ w,
                                 const long long* __restrict__ src,
                                 const long long* __restrict__ dst, int ne) {
  constexpr int CH = D / 4;  // power of two -> shifts
  long long i = (long long)blockIdx.x * blockDim.x + threadIdx.x;
  if (i >= (long long)ne * CH) return;
  int e = (int)(i / CH);
  int c = (int)(i - (long long)e * CH);
  int s = (int)src[e], d = (int)dst[e];
  float we = w[e];
  const float4* ps = (const float4*)(p + (size_t)s * D);
  float4 v = ps[c];
  float* o = out + (size_t)d * D + c * 4;
  atomicAdd(o + 0, v.x * we);
  atomicAdd(o + 1, v.y * we);
  atomicAdd(o + 2, v.z * we);
  atomicAdd(o + 3, v.w * we);
}

// ---------------- fused BN(eval) + ReLU, in place ----------------
__global__ void bn_relu_kernel(float* __restrict__ h,
                               const float* __restrict__ g,
                               const float* __restrict__ b,
                               const float* __restrict__ m,
                               const float* __restrict__ v, long long n) {
  long long i = (long long)blockIdx.x * blockDim.x + threadIdx.x;
  if (i >= n) return;
  int c = (int)(i & (DH - 1));
  float sc = g[c] * rsqrtf(v[c] + BN_EPSF);
  float val = (h[i] - m[c]) * sc + b[c];
  h[i] = fmaxf(val, 0.0f);
}

extern "C" void kernel_launch(void* const* d_in, const int* in_sizes, int n_in,
                              void* d_out, int out_size, void* d_ws,
                              size_t ws_size, hipStream_t stream) {
  const float* x = (const float*)d_in[0];
  const long long* ei = (const long long*)d_in[1];
  const long long* src = ei;             // edge_index[0, :]
  const long long* dst = ei + N_EDGES;   // edge_index[1, :]
  const float* Wn0 = (const float*)d_in[2];
  const float* cn0 = (const float*)d_in[3];
  const float* Wr0 = (const float*)d_in[4];
  const float* cr0 = (const float*)d_in[5];
  const float* Wn1 = (const float*)d_in[6];
  const float* cn1 = (const float*)d_in[7];
  const float* Wr1 = (const float*)d_in[8];
  const float* cr1 = (const float*)d_in[9];
  const float* Wn2 = (const float*)d_in[10];
  const float* cn2 = (const float*)d_in[11];
  const float* Wr2 = (const float*)d_in[12];
  const float* cr2 = (const float*)d_in[13];
  const float* g0 = (const float*)d_in[14];
  const float* b0 = (const float*)d_in[15];
  const float* m0 = (const float*)d_in[16];
  const float* v0 = (const float*)d_in[17];
  const float* g1 = (const float*)d_in[18];
  const float* b1 = (const float*)d_in[19];
  const float* m1 = (const float*)d_in[20];
  const float* v1 = (const float*)d_in[21];
  float* out = (float*)d_out;

  // workspace layout (floats)
  float* ws = (float*)d_ws;
  size_t o = 0;
  float* p  = ws + o; o += (size_t)N_NODES * DH;  // pre-transformed features
  float* hA = ws + o; o += (size_t)N_NODES * DH;  // activation buffer A
  float* hB = ws + o; o += (size_t)N_NODES * DH;  // activation buffer B
  float* w  = ws + o; o += (size_t)N_EDGES;       // edge weights (e then w)
  float* lg = ws + o; o += (size_t)N_EDGES;       // logits
  float* invn = ws + o; o += (size_t)N_NODES;     // 1/(||x||+eps)
  unsigned* mrep = (unsigned*)(ws + o); o += (size_t)N_NODES;  // seg-max (ord)
  float* denom = ws + o; o += (size_t)N_NODES;    // seg-sum

  const int B = 256;
  int nodeB = (N_NODES + B - 1) / B;
  int nodeWaveB = ((N_NODES * 32) + B - 1) / B;
  int edgeB = (N_EDGES + B - 1) / B;
  int edgeWaveB = (int)(((long long)N_EDGES * 32 + B - 1) / B);

  // ---- edge softmax weights ----
  init_seg_kernel<<<nodeB, B, 0, stream>>>(mrep, denom, N_NODES);
  row_invnorm_kernel<<<nodeWaveB, B, 0, stream>>>(x, invn, N_NODES);
  edge_logits_kernel<<<edgeWaveB, B, 0, stream>>>(x, src, dst, invn, lg, mrep,
                                                  N_EDGES);
  edge_exp_kernel<<<edgeB, B, 0, stream>>>(lg, dst, mrep, w, denom, N_EDGES);
  edge_norm_kernel<<<edgeB, B, 0, stream>>>(w, dst, denom, N_EDGES);

  // GEMM grids: one wave per 16x16 tile, 8 waves per block
  int tiles128 = (N_NODES / 16) * (DH / 16);    // 25000
  int tiles64  = (N_NODES / 16) * (DOUT / 16);  // 12500
  int gemmB128 = (tiles128 + 7) / 8;
  int gemmB64  = (tiles64 + 7) / 8;
  int aggB128 = (int)(((long long)N_EDGES * (DH / 4) + B - 1) / B);
  int aggB64  = (int)(((long long)N_EDGES * (DOUT / 4) + B - 1) / B);
  long long nAct = (long long)N_NODES * DH;
  int actB = (int)((nAct + B - 1) / B);

  // ---- layer 0: h0 = scatter((x@Wn0)[src]*w) + x@Wr0 + (cn0+cr0); BN+ReLU
  gemm_wmma_kernel<DIN, DH><<<gemmB128, B, 0, stream>>>(x, Wn0, nullptr,
                                                        nullptr, p, N_NODES);
  gemm_wmma_kernel<DIN, DH><<<gemmB128, B, 0, stream>>>(x, Wr0, cn0, cr0, hA,
                                                        N_NODES);
  aggregate_kernel<DH><<<aggB128, B, 0, stream>>>(hA, p, w, src, dst, N_EDGES);
  bn_relu_kernel<<<actB, B, 0, stream>>>(hA, g0, b0, m0, v0, nAct);

  // ---- layer 1 ----
  gemm_wmma_kernel<DH, DH><<<gemmB128, B, 0, stream>>>(hA, Wn1, nullptr,
                                                       nullptr, p, N_NODES);
  gemm_wmma_kernel<DH, DH><<<gemmB128, B, 0, stream>>>(hA, Wr1, cn1, cr1, hB,
                                                       N_NODES);
  aggregate_kernel<DH><<<aggB128, B, 0, stream>>>(hB, p, w, src, dst, N_EDGES);
  bn_relu_kernel<<<actB, B, 0, stream>>>(hB, g1, b1, m1, v1, nAct);

  // ---- layer 2 (pre-transform to 64 dims BEFORE aggregation) ----
  gemm_wmma_kernel<DH, DOUT><<<gemmB64, B, 0, stream>>>(hB, Wn2, nullptr,
                                                        nullptr, p, N_NODES);
  gemm_wmma_kernel<DH, DOUT><<<gemmB64, B, 0, stream>>>(hB, Wr2, cn2, cr2, out,
                                                        N_NODES);
  aggregate_kernel<DOUT><<<aggB64, B, 0, stream>>>(out, p, w, src, dst,
                                                   N_EDGES);
}